// DynScan_40261023432927
// MI455X (gfx1250) — compile-verified
//
#include <hip/hip_runtime.h>
#include <hip/hip_bf16.h>
#include <math.h>

#define BB 4
#define KK 32
#define NN 512
#define DD 64
#define LL 10
#define BK (BB*KK)
#define QKVSZ ((size_t)BK*NN*DD)
#define LN_EPS 1e-5f

typedef float v2f __attribute__((ext_vector_type(2)));
typedef float v8f __attribute__((ext_vector_type(8)));
typedef float f4  __attribute__((ext_vector_type(4)));

// ---------------------------------------------------------------------------
// Async global->LDS staging of a 64x64 f32 tile (16 KB) by 256 threads.
// Uses GLOBAL_LOAD_ASYNC_TO_LDS_B128 (ASYNCcnt) when the builtin exists,
// else coalesced float4 register copies.
// Builtin param 1 (per clang diagnostic) is `int __vector_size__(16)` in the
// global (__device__) address space; dst is the LDS (addrspace 3) pointer.
// ---------------------------------------------------------------------------
#if __has_builtin(__builtin_amdgcn_global_load_async_to_lds_b128)
#define HAS_ASYNC_LDS 1
typedef int v4i_vs __attribute__((vector_size(16)));
typedef __attribute__((address_space(1))) v4i_vs* g_v4i;
typedef __attribute__((address_space(3))) v4i_vs* l_v4i;
#else
#define HAS_ASYNC_LDS 0
#endif

__device__ __forceinline__ void stage_tile_4096(const float* __restrict__ g,
                                                float* s, int tid) {
#if HAS_ASYNC_LDS
#pragma unroll
  for (int i = 0; i < 4; ++i) {
    const size_t e = (size_t)(tid + 256*i) * 4;   // float index, 16B aligned
    __builtin_amdgcn_global_load_async_to_lds_b128(
        (g_v4i)(g + e), (l_v4i)(s + e), 0, 0);
  }
#else
#pragma unroll
  for (int i = 0; i < 4; ++i) {
    const int e = tid + 256*i;
    ((f4*)s)[e] = ((const f4*)g)[e];
  }
#endif
}

__device__ __forceinline__ void wait_stage() {
#if HAS_ASYNC_LDS
#if __has_builtin(__builtin_amdgcn_s_wait_asynccnt)
  __builtin_amdgcn_s_wait_asynccnt(0);
#else
  asm volatile("s_wait_asynccnt 0" ::: "memory");
#endif
#endif
}

// ---------------------------------------------------------------------------
// Wave-level f32 WMMA 16x16 tile GEMM on V_WMMA_F32_16X16X4_F32.
// A 16x4 : lanes 0-15 hold M=lane, v0=K0 v1=K1; lanes 16-31 hold K2/K3
// B 4x16 : lanes 0-15 hold N=lane, v0=K0 v1=K1; lanes 16-31 hold K2/K3
// C 16x16: v[i] holds (M = i + 8*(lane>>4), N = lane&15)
// ---------------------------------------------------------------------------
__device__ __forceinline__ v8f wmma_rr(const float* __restrict__ A, int lda,
                                       const float* __restrict__ B, int ldb,
                                       v8f acc, int kdim) {
  const int lane = threadIdx.x & 31;
  const int half = lane >> 4, lr = lane & 15;
  for (int kk = 0; kk < kdim; kk += 4) {
    const int ka = kk + 2*half;
    v2f a, b;
    a.x = A[lr*lda + ka];       a.y = A[lr*lda + ka + 1];
    b.x = B[ka*ldb + lr];       b.y = B[(ka+1)*ldb + lr];
    acc = __builtin_amdgcn_wmma_f32_16x16x4_f32(false, a, false, b,
                                                (short)0, acc, false, false);
  }
  return acc;
}

// B = M^T, M row-major [16 x kdim] pre-offset to the 16 cols
__device__ __forceinline__ v8f wmma_rt(const float* __restrict__ A, int lda,
                                       const float* __restrict__ M, int ldm,
                                       v8f acc, int kdim) {
  const int lane = threadIdx.x & 31;
  const int half = lane >> 4, lr = lane & 15;
  for (int kk = 0; kk < kdim; kk += 4) {
    const int ka = kk + 2*half;
    v2f a, b;
    a.x = A[lr*lda + ka];       a.y = A[lr*lda + ka + 1];
    b.x = M[lr*ldm + ka];       b.y = M[lr*ldm + ka + 1];
    acc = __builtin_amdgcn_wmma_f32_16x16x4_f32(false, a, false, b,
                                                (short)0, acc, false, false);
  }
  return acc;
}

// ---------------------------------------------------------------------------
// K0: Wq' = Wq @ W_att, bq' = bq @ W_att
// ---------------------------------------------------------------------------
__global__ __launch_bounds__(256) void k_fuse(
    const float* __restrict__ Wq, const float* __restrict__ bq,
    const float* __restrict__ Watt, float* __restrict__ Wqp,
    float* __restrict__ bqp) {
  const int t = threadIdx.x;
  for (int o = t; o < DD*DD; o += 256) {
    const int i = o / DD, j = o % DD;
    float s = 0.f;
    for (int k = 0; k < DD; ++k) s += Wq[i*DD + k] * Watt[k*DD + j];
    Wqp[o] = s;
  }
  if (t < DD) {
    float s = 0.f;
    for (int k = 0; k < DD; ++k) s += bq[k] * Watt[k*DD + t];
    bqp[t] = s;
  }
}

// ---------------------------------------------------------------------------
// K1: Q'/K/V projections. Block = 64 rows staged to LDS (async),
// 48 wave-tile tasks (4 row-tiles x 3 mats x 4 col-tiles). grid = BK*N/64.
// ---------------------------------------------------------------------------
__global__ __launch_bounds__(256) void k_proj(
    const float* __restrict__ H,
    const float* __restrict__ Wqp, const float* __restrict__ bqp,
    const float* __restrict__ Wk,  const float* __restrict__ bk,
    const float* __restrict__ Wv,  const float* __restrict__ bv,
    float* __restrict__ Qp, float* __restrict__ Kf, float* __restrict__ Vf) {
  __shared__ float s_x[64*DD];
  const int tid = threadIdx.x, wid = tid >> 5;
  const size_t r0 = (size_t)blockIdx.x * 64;
  stage_tile_4096(H + r0*DD, s_x, tid);
  wait_stage();
  __syncthreads();
  for (int tt = wid; tt < 48; tt += 8) {
    const int rl  = tt / 12;
    const int sub = tt % 12;
    const int mat = sub >> 2, ct = sub & 3;
    const float* W;  const float* bias;  float* out;
    if (mat == 0)      { W = Wqp; bias = bqp; out = Qp; }
    else if (mat == 1) { W = Wk;  bias = bk;  out = Kf; }
    else               { W = Wv;  bias = bv;  out = Vf; }
    v8f acc = {};
    acc = wmma_rr(s_x + rl*16*DD, DD, W + ct*16, DD, acc, DD);
    const int lane = tid & 31, half = lane >> 4, lr = lane & 15;
    const float bc = bias[ct*16 + lr];
    float* C = out + (r0 + rl*16)*DD + ct*16;
#pragma unroll
    for (int i = 0; i < 8; ++i) C[(i + 8*half)*DD + lr] = acc[i] + bc;
  }
}

// ---------------------------------------------------------------------------
// K2: masked graph attention + residual + LN, flash-style online softmax.
// adj is read exactly once, coalesced, packed into an LDS bitmask.
// Zero-degree rows (all-masked) resolve to out = V[row] in the epilogue.
// ---------------------------------------------------------------------------
__global__ __launch_bounds__(256) void k_mpattn(
    const float* __restrict__ H, const float* __restrict__ adj,
    const float* __restrict__ Qp, const float* __restrict__ Kf,
    const float* __restrict__ Vf,
    const float* __restrict__ gam, const float* __restrict__ bet,
    float* __restrict__ Hmp) {
  __shared__ float s_q[64*DD];
  __shared__ float s_s[64][64];
  __shared__ float s_o[64][DD];
  __shared__ unsigned s_mb[64][2];
  __shared__ float s_m[64], s_l[64], s_f[64];
  const int p  = blockIdx.x >> 3;
  const int r0 = (blockIdx.x & 7) * 64;
  const int tid = threadIdx.x;
  const int wid = tid >> 5;
  const size_t rowbase = (size_t)p * NN + r0;

  stage_tile_4096(Qp + rowbase*DD, s_q, tid);
  for (int o = tid; o < 64*DD; o += 256) (&s_o[0][0])[o] = 0.f;
  if (tid < 64) { s_m[tid] = -INFINITY; s_l[tid] = 0.f; }
  wait_stage();
  __syncthreads();

  for (int c0 = 0; c0 < NN; c0 += 64) {
    if (tid < 128) s_mb[tid >> 1][tid & 1] = 0u;
    // scores = Q' @ K^T * rsqrt(D)
    for (int tt = wid; tt < 16; tt += 8) {
      const int tm = tt & 3, tn = tt >> 2;
      v8f acc = {};
      acc = wmma_rt(s_q + tm*16*DD, DD,
                    Kf + ((size_t)p*NN + c0 + tn*16)*DD, DD, acc, DD);
      const int lane = tid & 31, half = lane >> 4, lr = lane & 15;
#pragma unroll
      for (int i = 0; i < 8; ++i)
        s_s[tm*16 + i + 8*half][tn*16 + lr] = acc[i] * 0.125f;
    }
    __syncthreads();
    // adj chunk: single coalesced pass -> LDS bitmask (ds_or)
    {
      const float* ab = adj + ((size_t)p*NN + r0)*NN + c0;
#pragma unroll
      for (int i = 0; i < 16; ++i) {
        const int e = tid + 256*i;           // 0..4095
        const int rr = e >> 6, cc = e & 63;
        if (ab[(size_t)rr*NN + cc] != 0.f)
          atomicOr(&s_mb[rr][cc >> 5], 1u << (cc & 31));
      }
    }
    __syncthreads();
    // per-row mask + online softmax update
    if (tid < 64) {
      const unsigned m0 = s_mb[tid][0], m1 = s_mb[tid][1];
      float mloc = -INFINITY;
      for (int c = 0; c < 64; ++c) {
        const bool keep = (c < 32) ? ((m0 >> c) & 1u) : ((m1 >> (c-32)) & 1u);
        const float sv = keep ? s_s[tid][c] : -INFINITY;
        s_s[tid][c] = sv;
        mloc = fmaxf(mloc, sv);
      }
      const float mold = s_m[tid];
      const float mnew = fmaxf(mold, mloc);
      float f = 1.f, lsum = 0.f;
      if (mnew != -INFINITY) {
        f = (mold == -INFINITY) ? 0.f : __expf(mold - mnew);
        for (int c = 0; c < 64; ++c) {
          const float sv = s_s[tid][c];
          const float pv = (sv == -INFINITY) ? 0.f : __expf(sv - mnew);
          s_s[tid][c] = pv; lsum += pv;
        }
      } else {
        for (int c = 0; c < 64; ++c) s_s[tid][c] = 0.f;
      }
      s_m[tid] = mnew;
      s_l[tid] = s_l[tid]*f + lsum;
      s_f[tid] = f;
    }
    __syncthreads();
    for (int o = tid; o < 64*DD; o += 256) s_o[o >> 6][o & 63] *= s_f[o >> 6];
    __syncthreads();
    // O += P @ Vchunk
    for (int tt = wid; tt < 16; tt += 8) {
      const int tm = tt & 3, tn = tt >> 2;
      const int lane = tid & 31, half = lane >> 4, lr = lane & 15;
      v8f acc;
#pragma unroll
      for (int i = 0; i < 8; ++i) acc[i] = s_o[tm*16 + i + 8*half][tn*16 + lr];
      acc = wmma_rr(&s_s[tm*16][0], 64,
                    Vf + ((size_t)p*NN + c0)*DD + tn*16, DD, acc, DD);
#pragma unroll
      for (int i = 0; i < 8; ++i) s_o[tm*16 + i + 8*half][tn*16 + lr] = acc[i];
    }
    __syncthreads();
  }
  // residual + LayerNorm (l==0 rows: reference self-loop => out = V[row])
  if (tid < 64) {
    const float* xr = H + (rowbase + tid)*DD;
    float mean = 0.f;
    if (s_l[tid] > 0.f) {
      const float invl = 1.f/s_l[tid];
      for (int c = 0; c < DD; ++c) {
        const float y = xr[c] + s_o[tid][c]*invl;
        s_s[tid][c] = y; mean += y;
      }
    } else {
      const float* vr = Vf + (rowbase + tid)*DD;
      for (int c = 0; c < DD; ++c) {
        const float y = xr[c] + vr[c];
        s_s[tid][c] = y; mean += y;
      }
    }
    mean *= (1.f/DD);
    float var = 0.f;
    for (int c = 0; c < DD; ++c) { const float d = s_s[tid][c]-mean; var += d*d; }
    var *= (1.f/DD);
    const float rstd = rsqrtf(var + LN_EPS);
    float* outr = Hmp + (rowbase + tid)*DD;
    for (int c = 0; c < DD; ++c)
      outr[c] = (s_s[tid][c] - mean)*rstd*gam[c] + bet[c];
  }
}

// ---------------------------------------------------------------------------
// slots0 = mu + exp(log_sigma) * eps
// ---------------------------------------------------------------------------
__global__ __launch_bounds__(256) void k_slotinit(
    const float* __restrict__ mu, const float* __restrict__ lsig,
    const float* __restrict__ eps, float* __restrict__ slots) {
  for (int o = threadIdx.x + blockIdx.x*256; o < BB*LL*DD; o += 256*gridDim.x) {
    const int ld = o % (LL*DD);
    slots[o] = mu[ld] + __expf(lsig[ld]) * eps[o];
  }
}

// ---------------------------------------------------------------------------
// K3: per-timestep slot K/V projection. Block = 64 rows staged to LDS,
// 32 wave-tile tasks. grid = B*N/64 = 32.
// ---------------------------------------------------------------------------
__global__ __launch_bounds__(256) void k_slotkv(
    const float* __restrict__ Hmp, int t,
    const float* __restrict__ Wk, const float* __restrict__ bk,
    const float* __restrict__ Wv, const float* __restrict__ bv,
    float* __restrict__ Kout, float* __restrict__ Vout) {
  __shared__ float s_x[64*DD];
  const int tid = threadIdx.x, wid = tid >> 5;
  const int r0 = blockIdx.x * 64;              // flat row in [0, B*N)
  const int b  = r0 / NN, n0 = r0 % NN;
  stage_tile_4096(Hmp + (((size_t)b*KK + t)*NN + n0)*DD, s_x, tid);
  wait_stage();
  __syncthreads();
  for (int tt = wid; tt < 32; tt += 8) {
    const int rl  = tt >> 3;
    const int sub = tt & 7;
    const int mat = sub >> 2, ct = sub & 3;
    const float* W    = mat ? Wv : Wk;
    const float* bias = mat ? bv : bk;
    float* out = (mat ? Vout : Kout) + ((size_t)r0 + rl*16)*DD + ct*16;
    v8f acc = {};
    acc = wmma_rr(s_x + rl*16*DD, DD, W + ct*16, DD, acc, DD);
    const int lane = tid & 31, half = lane >> 4, lr = lane & 15;
    const float bc = bias[ct*16 + lr];
#pragma unroll
    for (int i = 0; i < 8; ++i) out[(i + 8*half)*DD + lr] = acc[i] + bc;
  }
}

// ---------------------------------------------------------------------------
// K4: one slot-attention + GRU + FFN + LN step (40 slot rows -> VALU).
// ---------------------------------------------------------------------------
__global__ __launch_bounds__(256) void k_slotstep(
    int t, const float* __restrict__ Kin, const float* __restrict__ Vin,
    const float* __restrict__ Wq,  const float* __restrict__ bq,
    const float* __restrict__ Wih, const float* __restrict__ bih,
    const float* __restrict__ Whh, const float* __restrict__ bhh,
    const float* __restrict__ W1,  const float* __restrict__ b1,
    const float* __restrict__ W2,  const float* __restrict__ b2,
    const float* __restrict__ gam, const float* __restrict__ bet,
    float* __restrict__ slots, float* __restrict__ S) {
  __shared__ float s_sl[LL][DD];
  __shared__ float s_q [LL][DD];
  __shared__ float s_at[LL][NN];
  __shared__ float s_up[LL][DD];
  __shared__ float s_gi[LL][3*DD];
  __shared__ float s_gh[LL][3*DD];
  __shared__ float s_s2[LL][DD];
  __shared__ float s_t1[LL][DD];
  const int b = blockIdx.x;
  const int tid = threadIdx.x;
  const float* slb = slots + (size_t)b*LL*DD;
  for (int o = tid; o < LL*DD; o += 256) s_sl[o/DD][o%DD] = slb[o];
  __syncthreads();
  for (int o = tid; o < LL*DD; o += 256) {
    const int l = o/DD, j = o%DD;
    float s = bq[j];
    for (int k = 0; k < DD; ++k) s += s_sl[l][k]*Wq[k*DD + j];
    s_q[l][j] = s;
  }
  __syncthreads();
  for (int o = tid; o < LL*NN; o += 256) {
    const int l = o/NN, n = o%NN;
    const float* kr = Kin + ((size_t)b*NN + n)*DD;
    float s = 0.f;
    for (int k = 0; k < DD; ++k) s += s_q[l][k]*kr[k];
    s_at[l][n] = s * 0.125f;
  }
  __syncthreads();
  if (tid < LL) {
    float m = -INFINITY;
    for (int n = 0; n < NN; ++n) m = fmaxf(m, s_at[tid][n]);
    float sum = 0.f;
    for (int n = 0; n < NN; ++n) { const float e = __expf(s_at[tid][n]-m); s_at[tid][n] = e; sum += e; }
    const float inv = 1.f/sum;
    for (int n = 0; n < NN; ++n) s_at[tid][n] *= inv;
  }
  __syncthreads();
  for (int o = tid; o < LL*DD; o += 256) {
    const int l = o/DD, j = o%DD;
    float s = 0.f;
    for (int n = 0; n < NN; ++n) s += s_at[l][n] * Vin[((size_t)b*NN + n)*DD + j];
    s_up[l][j] = s;
  }
  __syncthreads();
  for (int o = tid; o < LL*3*DD; o += 256) {
    const int l = o/(3*DD), j = o%(3*DD);
    float si = bih[j], sh = bhh[j];
    for (int k = 0; k < DD; ++k) {
      si += s_up[l][k]*Wih[k*3*DD + j];
      sh += s_sl[l][k]*Whh[k*3*DD + j];
    }
    s_gi[l][j] = si; s_gh[l][j] = sh;
  }
  __syncthreads();
  for (int o = tid; o < LL*DD; o += 256) {
    const int l = o/DD, j = o%DD;
    const float r = 1.f/(1.f + __expf(-(s_gi[l][j]      + s_gh[l][j])));
    const float z = 1.f/(1.f + __expf(-(s_gi[l][DD+j]   + s_gh[l][DD+j])));
    const float n = tanhf(      s_gi[l][2*DD+j] + r*s_gh[l][2*DD+j]);
    s_s2[l][j] = (1.f - z)*n + z*s_sl[l][j];
  }
  __syncthreads();
  for (int o = tid; o < LL*DD; o += 256) {
    const int l = o/DD, j = o%DD;
    float s = b1[j];
    for (int k = 0; k < DD; ++k) s += s_s2[l][k]*W1[k*DD + j];
    s_t1[l][j] = fmaxf(s, 0.f);
  }
  __syncthreads();
  for (int o = tid; o < LL*DD; o += 256) {
    const int l = o/DD, j = o%DD;
    float s = b2[j];
    for (int k = 0; k < DD; ++k) s += s_t1[l][k]*W2[k*DD + j];
    s_up[l][j] = s_s2[l][j] + s;       // pre-LN
  }
  __syncthreads();
  if (tid < LL) {
    float mean = 0.f;
    for (int j = 0; j < DD; ++j) mean += s_up[tid][j];
    mean *= (1.f/DD);
    float var = 0.f;
    for (int j = 0; j < DD; ++j) { const float d = s_up[tid][j]-mean; var += d*d; }
    var *= (1.f/DD);
    const float rstd = rsqrtf(var + LN_EPS);
    float* so = slots + ((size_t)b*LL + tid)*DD;
    float* So = S + (((size_t)b*KK + t)*LL + tid)*DD;
    for (int j = 0; j < DD; ++j) {
      const float y = (s_up[tid][j]-mean)*rstd*gam[j] + bet[j];
      so[j] = y; So[j] = y;
    }
  }
}

// ---------------------------------------------------------------------------
// K5: decoder. Block = (pair p, 64-row tile). In-place over d_out; the
// async h-tile DMA overlaps the VALU slot-K/V projection.
// ---------------------------------------------------------------------------
__global__ __launch_bounds__(256) void k_dec(
    const float* __restrict__ S,
    const float* __restrict__ Wq, const float* __restrict__ bq,
    const float* __restrict__ Wk, const float* __restrict__ bk,
    const float* __restrict__ Wv, const float* __restrict__ bv,
    const float* __restrict__ W1, const float* __restrict__ b1,
    const float* __restrict__ W2, const float* __restrict__ b2,
    const float* __restrict__ gam, const float* __restrict__ bet,
    float* __restrict__ HZ) {
  __shared__ float s_h[64][DD];
  __shared__ float s_z[64][DD];
  __shared__ float s_t[64][DD];
  __shared__ float s_k[LL][DD];
  __shared__ float s_v[LL][DD];
  const int p  = blockIdx.x >> 3;
  const int r0 = (blockIdx.x & 7) * 64;
  const int tid = threadIdx.x;
  const int wid = tid >> 5;
  const size_t rowbase = (size_t)p*NN + r0;

  stage_tile_4096(HZ + rowbase*DD, &s_h[0][0], tid);   // overlaps KV below
  const float* Sb = S + (size_t)p*LL*DD;
  for (int o = tid; o < 2*LL*DD; o += 256) {
    const int mat = o / (LL*DD);
    const int l = (o % (LL*DD)) / DD, j = o % DD;
    const float* W = mat ? Wv : Wk;
    float s = (mat ? bv : bk)[j];
    for (int k = 0; k < DD; ++k) s += Sb[l*DD + k]*W[k*DD + j];
    (mat ? s_v : s_k)[l][j] = s;
  }
  wait_stage();
  __syncthreads();
  // Q = h @ Wq + bq
  for (int tt = wid; tt < 16; tt += 8) {
    const int tm = tt & 3, tn = tt >> 2;
    v8f acc = {};
    acc = wmma_rr(&s_h[tm*16][0], DD, Wq + tn*16, DD, acc, DD);
    const int lane = tid & 31, half = lane >> 4, lr = lane & 15;
    const float bc = bq[tn*16 + lr];
#pragma unroll
    for (int i = 0; i < 8; ++i)
      s_z[tm*16 + i + 8*half][tn*16 + lr] = acc[i] + bc;
  }
  __syncthreads();
  // scores over 10 slots -> softmax -> z row (overwrite s_z row in place)
  if (tid < 64) {
    float sc[LL];
    float m = -INFINITY;
    for (int l = 0; l < LL; ++l) {
      float s = 0.f;
      for (int k = 0; k < DD; ++k) s += s_z[tid][k]*s_k[l][k];
      sc[l] = s*0.125f; m = fmaxf(m, sc[l]);
    }
    float sum = 0.f;
    for (int l = 0; l < LL; ++l) { sc[l] = __expf(sc[l]-m); sum += sc[l]; }
    const float inv = 1.f/sum;
    for (int j = 0; j < DD; ++j) s_z[tid][j] = 0.f;
    for (int l = 0; l < LL; ++l) {
      const float a = sc[l]*inv;
      for (int j = 0; j < DD; ++j) s_z[tid][j] += a*s_v[l][j];
    }
  }
  __syncthreads();
  // t1 = relu(z@W1 + b1)
  for (int tt = wid; tt < 16; tt += 8) {
    const int tm = tt & 3, tn = tt >> 2;
    v8f acc = {};
    acc = wmma_rr(&s_z[tm*16][0], DD, W1 + tn*16, DD, acc, DD);
    const int lane = tid & 31, half = lane >> 4, lr = lane & 15;
    const float bc = b1[tn*16 + lr];
#pragma unroll
    for (int i = 0; i < 8; ++i)
      s_t[tm*16 + i + 8*half][tn*16 + lr] = fmaxf(acc[i] + bc, 0.f);
  }
  __syncthreads();
  // f = t1@W2 + b2  -> s_z
  for (int tt = wid; tt < 16; tt += 8) {
    const int tm = tt & 3, tn = tt >> 2;
    v8f acc = {};
    acc = wmma_rr(&s_t[tm*16][0], DD, W2 + tn*16, DD, acc, DD);
    const int lane = tid & 31, half = lane >> 4, lr = lane & 15;
    const float bc = b2[tn*16 + lr];
#pragma unroll
    for (int i = 0; i < 8; ++i)
      s_z[tm*16 + i + 8*half][tn*16 + lr] = acc[i] + bc;
  }
  __syncthreads();
  // Z = LN(h + f)
  if (tid < 64) {
    float mean = 0.f;
    for (int j = 0; j < DD; ++j) {
      const float y = s_h[tid][j] + s_z[tid][j];
      s_t[tid][j] = y; mean += y;
    }
    mean *= (1.f/DD);
    float var = 0.f;
    for (int j = 0; j < DD; ++j) { const float d = s_t[tid][j]-mean; var += d*d; }
    var *= (1.f/DD);
    const float rstd = rsqrtf(var + LN_EPS);
    float* outr = HZ + (rowbase + tid)*DD;
    for (int j = 0; j < DD; ++j)
      outr[j] = (s_t[tid][j] - mean)*rstd*gam[j] + bet[j];
  }
}

// ---------------------------------------------------------------------------
extern "C" void kernel_launch(void* const* d_in, const int* in_sizes, int n_in,
                              void* d_out, int out_size, void* d_ws, size_t ws_size,
                              hipStream_t stream) {
  (void)in_sizes; (void)n_in; (void)out_size; (void)ws_size;
  const float* H    = (const float*)d_in[0];
  const float* adj  = (const float*)d_in[1];
  const float* eps  = (const float*)d_in[2];
  const float* mWq  = (const float*)d_in[3];
  const float* mbq  = (const float*)d_in[4];
  const float* mWk  = (const float*)d_in[5];
  const float* mbk  = (const float*)d_in[6];
  const float* mWv  = (const float*)d_in[7];
  const float* mbv  = (const float*)d_in[8];
  const float* mWatt= (const float*)d_in[9];
  const float* mg   = (const float*)d_in[10];
  const float* mb   = (const float*)d_in[11];
  const float* smu  = (const float*)d_in[12];
  const float* slog = (const float*)d_in[13];
  const float* sWq  = (const float*)d_in[14];
  const float* sbq  = (const float*)d_in[15];
  const float* sWk  = (const float*)d_in[16];
  const float* sbk  = (const float*)d_in[17];
  const float* sWv  = (const float*)d_in[18];
  const float* sbv  = (const float*)d_in[19];
  const float* sWih = (const float*)d_in[20];
  const float* sbih = (const float*)d_in[21];
  const float* sWhh = (const float*)d_in[22];
  const float* sbhh = (const float*)d_in[23];
  const float* sW1  = (const float*)d_in[24];
  const float* sb1  = (const float*)d_in[25];
  const float* sW2  = (const float*)d_in[26];
  const float* sb2  = (const float*)d_in[27];
  const float* sg   = (const float*)d_in[28];
  const float* sb   = (const float*)d_in[29];
  const float* dWq  = (const float*)d_in[30];
  const float* dbq  = (const float*)d_in[31];
  const float* dWk  = (const float*)d_in[32];
  const float* dbk  = (const float*)d_in[33];
  const float* dWv  = (const float*)d_in[34];
  const float* dbv  = (const float*)d_in[35];
  const float* dW1  = (const float*)d_in[36];
  const float* db1  = (const float*)d_in[37];
  const float* dW2  = (const float*)d_in[38];
  const float* db2  = (const float*)d_in[39];
  const float* dg   = (const float*)d_in[40];
  const float* db   = (const float*)d_in[41];

  float* ws    = (float*)d_ws;               // ~52 MB fp32 scratch
  float* Wqp   = ws;                         // 4096
  float* bqp   = ws + 4096;                  // 64 (padded to 4160)
  float* Qp    = ws + 4160;                  // BK*N*D
  float* Kf    = Qp + QKVSZ;
  float* Vf    = Kf + QKVSZ;
  float* sKbuf = Vf + QKVSZ;                 // B*N*D
  float* sVbuf = sKbuf + (size_t)BB*NN*DD;
  float* slots = sVbuf + (size_t)BB*NN*DD;   // B*L*D
  float* Sbuf  = slots + (size_t)BB*LL*DD;   // B*K*L*D
  float* Z     = (float*)d_out;              // Hmp lives here between phases

  k_fuse    <<<1,    256, 0, stream>>>(mWq, mbq, mWatt, Wqp, bqp);
  k_proj    <<<1024, 256, 0, stream>>>(H, Wqp, bqp, mWk, mbk, mWv, mbv, Qp, Kf, Vf);
  k_mpattn  <<<1024, 256, 0, stream>>>(H, adj, Qp, Kf, Vf, mg, mb, Z);
  k_slotinit<<<4,    256, 0, stream>>>(smu, slog, eps, slots);
  for (int t = 0; t < KK; ++t) {
    k_slotkv  <<<32, 256, 0, stream>>>(Z, t, sWk, sbk, sWv, sbv, sKbuf, sVbuf);
    k_slotstep<<<BB, 256, 0, stream>>>(t, sKbuf, sVbuf, sWq, sbq, sWih, sbih,
                                       sWhh, sbhh, sW1, sb1, sW2, sb2, sg, sb,
                                       slots, Sbuf);
  }
  k_dec     <<<1024, 256, 0, stream>>>(Sbuf, dWq, dbq, dWk, dbk, dWv, dbv,
                                       dW1, db1, dW2, db2, dg, db, Z);
}